// attention_edge_to_node_19258633355662
// MI455X (gfx1250) — compile-verified
//
#include <hip/hip_runtime.h>
#include <hip/hip_bf16.h>

typedef __attribute__((ext_vector_type(2))) float v2f;
typedef __attribute__((ext_vector_type(4))) float f4;
typedef __attribute__((ext_vector_type(8))) float v8f;

#define B_  2
#define N_  256
#define D_  256
#define DH_ 256
#define ROWS_ (B_ * N_)   // 512 rows of x / K / V / Kq

// ---------------------------------------------------------------------------
// Kernel A: K = x @ Wk + bk ; V = x @ Wv + bv
// One wave per 16x16 output tile, V_WMMA_F32_16X16X4_F32 over K-dim in steps
// of 4. Waves 0..511 -> K tiles, waves 512..1023 -> V tiles.
// A fragment (16x4 f32, ISA layout): lane half = K[1], VGPR j = K[0].
// B fragment (4x16 f32, mirrored):   lane half = K[1], VGPR j = K[0].
// C/D: VGPR r, lanes 0-15 -> M=r, lanes 16-31 -> M=r+8; N = lane&15.
// ---------------------------------------------------------------------------
__global__ __launch_bounds__(256) void kv_gemm(
    const float* __restrict__ x,
    const float* __restrict__ Wk, const float* __restrict__ bk,
    const float* __restrict__ Wv, const float* __restrict__ bv,
    float* __restrict__ Kout, float* __restrict__ Vout)
{
    int wave = (blockIdx.x * blockDim.x + threadIdx.x) >> 5;   // 0..1023
    int lane = threadIdx.x & 31;
    int half = lane >> 4;       // K high bit
    int lm   = lane & 15;       // M (for A) / N (for B)

    int which = wave >> 9;      // 0 = K, 1 = V
    int tile  = wave & 511;     // 32 m-tiles x 16 n-tiles
    int mBase = (tile >> 4) * 16;
    int nBase = (tile & 15) * 16;

    const float* W    = which ? Wv : Wk;
    const float* bias = which ? bv : bk;
    float*       Out  = which ? Vout : Kout;

    const float* xrow = x + (size_t)(mBase + lm) * D_;
    const int    nCol = nBase + lm;

    v8f acc = {};
    for (int k0 = 0; k0 < D_; k0 += 4) {
        const int ka = k0 + 2 * half;
        v2f a, b;
        a.x = xrow[ka];
        a.y = xrow[ka + 1];
        b.x = W[(size_t)ka       * DH_ + nCol];
        b.y = W[(size_t)(ka + 1) * DH_ + nCol];
        acc = __builtin_amdgcn_wmma_f32_16x16x4_f32(
            false, a, false, b, (short)0, acc, false, false);
    }

    const float bb = bias[nCol];
#pragma unroll
    for (int r = 0; r < 8; ++r) {
        int m = mBase + r + 8 * half;
        Out[(size_t)m * DH_ + nCol] = acc[r] + bb;
    }
}

// ---------------------------------------------------------------------------
// Kernel B: Kq = K @ Wq^T  (Kq[row,d] = sum_h K[row,h] * Wq[d,h])
//           cq[row] = dot(bq, K[row,:])
// Waves 0..511: WMMA tiles (B fragment reads Wq rows contiguously).
// Waves 512..1023: one wave per cq row, wave32 xor-shuffle reduction.
// ---------------------------------------------------------------------------
__global__ __launch_bounds__(256) void kq_gemm(
    const float* __restrict__ Kmat, const float* __restrict__ Wq,
    const float* __restrict__ bq,
    float* __restrict__ Kq, float* __restrict__ cq)
{
    int wave = (blockIdx.x * blockDim.x + threadIdx.x) >> 5;   // 0..1023
    int lane = threadIdx.x & 31;

    if (wave < 512) {
        int half  = lane >> 4;
        int lm    = lane & 15;
        int mBase = (wave >> 4) * 16;
        int nBase = (wave & 15) * 16;

        const float* arow = Kmat + (size_t)(mBase + lm) * DH_;   // K row (h-contig)
        const float* wrow = Wq   + (size_t)(nBase + lm) * DH_;   // Wq row d (h-contig)

        v8f acc = {};
        for (int k0 = 0; k0 < DH_; k0 += 4) {
            const int ka = k0 + 2 * half;
            v2f a, b;
            a.x = arow[ka]; a.y = arow[ka + 1];
            b.x = wrow[ka]; b.y = wrow[ka + 1];
            acc = __builtin_amdgcn_wmma_f32_16x16x4_f32(
                false, a, false, b, (short)0, acc, false, false);
        }
#pragma unroll
        for (int r = 0; r < 8; ++r) {
            int m = mBase + r + 8 * half;
            Kq[(size_t)m * D_ + nBase + lm] = acc[r];
        }
    } else {
        int row = wave - 512;                 // 0..511
        const float* krow = Kmat + (size_t)row * DH_;
        float s = 0.f;
#pragma unroll
        for (int t = 0; t < 8; ++t) {
            int h = t * 32 + lane;
            s += bq[h] * krow[h];
        }
#pragma unroll
        for (int off = 16; off; off >>= 1)
            s += __shfl_xor(s, off, 32);
        if (lane == 0) cq[row] = s;
    }
}

// ---------------------------------------------------------------------------
// Kernel C: bandwidth-bound stream over e.
// logit_i = scale*(e[b,j,i,:].Kq[b,i,:] + cq[b,i])
// logit_j = scale*(e[b,j,i,:].Kq[b,j,:] + cq[b,j])
// ai = sigmoid(logit_i - logit_j);  e_out = ai*V[b,i,:] + (1-ai)*V[b,j,:]
// One block per (b,j, tile of 32 i-rows); one wave per i-row per pass.
// e reads / e_out writes use non-temporal hints (keep L2 for Kq/V).
// ---------------------------------------------------------------------------
__global__ __launch_bounds__(256) void edge_out(
    const float* __restrict__ e,
    const float* __restrict__ Kq, const float* __restrict__ V,
    const float* __restrict__ cq,
    float* __restrict__ eout)
{
    __shared__ float sKqj[D_];
    __shared__ float sVj[D_];

    const int T    = N_ / 32;                 // 8 i-tiles per (b,j)
    const int bj   = blockIdx.x / T;          // row index for j: b*N + j
    const int tile = blockIdx.x % T;
    const int b    = bj >> 8;
    const int tid  = threadIdx.x;

    sKqj[tid] = Kq[(size_t)bj * D_ + tid];
    sVj[tid]  = V [(size_t)bj * D_ + tid];
    __syncthreads();

    const float cqj   = cq[bj];
    const float scale = 0.0625f;              // 1/sqrt(256)
    const int wave = tid >> 5;
    const int lane = tid & 31;

    for (int r = 0; r < 4; ++r) {
        const int i  = tile * 32 + r * 8 + wave;
        const int bi = b * N_ + i;

        const f4* erow = (const f4*)(e   + ((size_t)bj * N_ + i) * D_) + lane * 2;
        const f4* kqi  = (const f4*)(Kq  + (size_t)bi * D_)            + lane * 2;

        if (r < 3) __builtin_prefetch(erow + 8 * (D_ / 4) * 2, 0, 1);

        f4 e0 = __builtin_nontemporal_load(erow);
        f4 e1 = __builtin_nontemporal_load(erow + 1);
        f4 k0 = kqi[0];
        f4 k1 = kqi[1];
        const float* kqj = sKqj + lane * 8;

        float di = e0.x * k0.x + e0.y * k0.y + e0.z * k0.z + e0.w * k0.w
                 + e1.x * k1.x + e1.y * k1.y + e1.z * k1.z + e1.w * k1.w;
        float dj = e0.x * kqj[0] + e0.y * kqj[1] + e0.z * kqj[2] + e0.w * kqj[3]
                 + e1.x * kqj[4] + e1.y * kqj[5] + e1.z * kqj[6] + e1.w * kqj[7];

#pragma unroll
        for (int off = 16; off; off >>= 1) {
            di += __shfl_xor(di, off, 32);
            dj += __shfl_xor(dj, off, 32);
        }

        const float li = scale * (di + cq[bi]);
        const float lj = scale * (dj + cqj);
        const float ai = 1.0f / (1.0f + __expf(lj - li));
        const float aj = 1.0f - ai;

        const f4* vi = (const f4*)(V + (size_t)bi * D_) + lane * 2;
        f4 v0 = vi[0];
        f4 v1 = vi[1];
        const float* vj = sVj + lane * 8;

        f4 o0, o1;
        o0.x = ai * v0.x + aj * vj[0];
        o0.y = ai * v0.y + aj * vj[1];
        o0.z = ai * v0.z + aj * vj[2];
        o0.w = ai * v0.w + aj * vj[3];
        o1.x = ai * v1.x + aj * vj[4];
        o1.y = ai * v1.y + aj * vj[5];
        o1.z = ai * v1.z + aj * vj[6];
        o1.w = ai * v1.w + aj * vj[7];

        f4* orow = (f4*)(eout + ((size_t)bj * N_ + i) * D_) + lane * 2;
        __builtin_nontemporal_store(o0, orow);
        __builtin_nontemporal_store(o1, orow + 1);
    }
}

// ---------------------------------------------------------------------------
extern "C" void kernel_launch(void* const* d_in, const int* in_sizes, int n_in,
                              void* d_out, int out_size, void* d_ws, size_t ws_size,
                              hipStream_t stream)
{
    const float* x  = (const float*)d_in[0];
    const float* e  = (const float*)d_in[1];
    const float* Wq = (const float*)d_in[2];
    const float* bq = (const float*)d_in[3];
    const float* Wk = (const float*)d_in[4];
    const float* bk = (const float*)d_in[5];
    const float* Wv = (const float*)d_in[6];
    const float* bv = (const float*)d_in[7];

    float* ws   = (float*)d_ws;
    float* Kmat = ws;                       // 512*256
    float* Vmat = ws + 131072;              // 512*256
    float* KqB  = ws + 262144;              // 512*256
    float* cqB  = ws + 393216;              // 512

    float* out_x = (float*)d_out;                 // x passthrough (first tuple member)
    float* eoutp = out_x + (size_t)B_ * N_ * D_;  // e_out

    // tuple element 0: x (unchanged)
    hipMemcpyAsync(out_x, x, sizeof(float) * (size_t)B_ * N_ * D_,
                   hipMemcpyDeviceToDevice, stream);

    // 1024 waves = 128 blocks x 256 threads for both GEMM kernels
    kv_gemm<<<128, 256, 0, stream>>>(x, Wk, bk, Wv, bv, Kmat, Vmat);
    kq_gemm<<<128, 256, 0, stream>>>(Kmat, Wq, bq, KqB, cqB);

    // stream kernel: 512 (b,j) pairs * 8 i-tiles
    edge_out<<<B_ * N_ * (N_ / 32), 256, 0, stream>>>(e, KqB, Vmat, cqB, eoutp);
}